// CoordinateDescent_86053964742943
// MI455X (gfx1250) — compile-verified
//
#include <hip/hip_runtime.h>

typedef float v2f __attribute__((ext_vector_type(2)));
typedef float v8f __attribute__((ext_vector_type(8)));

// Problem shape from reference setup_inputs()
constexpr int B_ = 8;
constexpr int M_ = 2048;
constexpr int N_ = 2048;
constexpr int R_ = 32;

constexpr int TM  = 128;       // output rows per workgroup
constexpr int TK  = 32;        // K chunk staged in LDS (double buffered)
constexpr int TKP = TK + 4;    // padded row pitch (floats)

__device__ __forceinline__ void wait_async0() {
#if __has_builtin(__builtin_amdgcn_s_wait_asynccnt)
    __builtin_amdgcn_s_wait_asynccnt(0);
#else
    asm volatile("s_wait_asynccnt 0x0" ::: "memory");
#endif
}

// CDNA5 async copy: 16 bytes global -> LDS per lane, tracked by ASYNCcnt.
__device__ __forceinline__ void async_copy_b128(void* lds_ptr, const float* gptr) {
    const unsigned lds_off = (unsigned)(uintptr_t)lds_ptr;           // low 32 bits = LDS offset
    asm volatile("global_load_async_to_lds_b128 %0, %1, off"
                 :: "v"(lds_off), "v"((unsigned long long)(uintptr_t)gptr)
                 : "memory");
}

// ---------------------------------------------------------------------------
// Gram: G[b] = W[b]^T W[b],  W: [B, Kdim, 32] -> G: [B, 32, 32]
// ---------------------------------------------------------------------------
__global__ __launch_bounds__(1024)
void gram32_kernel(const float* __restrict__ W, float* __restrict__ G, int Kdim)
{
    __shared__ float lds[32 * R_];
    const int b = blockIdx.x;
    const int t = threadIdx.x;          // 0..1023 -> (r,s)
    const int r = t >> 5;
    const int s = t & 31;
    const float* Wb = W + (size_t)b * Kdim * R_;

    float acc = 0.f;
    for (int n0 = 0; n0 < Kdim; n0 += 32) {
        lds[t] = Wb[(size_t)(n0 + (t >> 5)) * R_ + (t & 31)];
        __syncthreads();
#pragma unroll
        for (int j = 0; j < 32; ++j)
            acc += lds[j * 32 + r] * lds[j * 32 + s];
        __syncthreads();
    }
    G[(size_t)b * R_ * R_ + t] = acc;
}

// ---------------------------------------------------------------------------
// GEMM: Out[b, i, r] = sum_k op(X)[i,k] * W[b, k, r]
//   TRANSX=0: op(X)[i,k] = X[b, i, k]   (async b128 staging, contiguous)
//   TRANSX=1: op(X)[i,k] = X[b, k, i]   (coalesced read + LDS scatter-transpose)
// Block = 256 threads = 8 waves (wave32). Each wave owns a 16x32 output slab
// (two v_wmma_f32_16x16x4_f32 accumulators). LDS is double-buffered so the
// next K chunk's fill overlaps the current chunk's WMMAs.
// LDS layout is fragment-friendly: both A and B tiles stored with K
// contiguous per row, so every WMMA operand fetch is one aligned ds_load_b64.
// ---------------------------------------------------------------------------
template <int TRANSX>
__global__ __launch_bounds__(256)
void gemm_xw_kernel(const float* __restrict__ X, const float* __restrict__ W,
                    float* __restrict__ Out, int Mdim, int Kdim)
{
    __shared__ float ldsX [2][TM][TKP];   // [m][k], 2 x 18 KB
    __shared__ float ldsWt[2][R_][TKP];   // [r][k] (transposed), 2 x 4.5 KB

    const int b    = blockIdx.y;
    const int m0   = blockIdx.x * TM;
    const int tid  = threadIdx.x;
    const int lane = tid & 31;
    const int wave = tid >> 5;        // 0..7
    const int wr   = wave * 16;       // wave's row offset in tile
    const int half = lane >> 4;       // 0 or 1
    const int l16  = lane & 15;

    const float* Xb = X + (size_t)b * Mdim * Kdim;   // M_*N_ either way
    const float* Wb = W + (size_t)b * Kdim * R_;

    v8f c0 = {};
    v8f c1 = {};

    // ---- tile staging ----------------------------------------------------
    auto stageX = [&](int k0, int buf) {
        if (TRANSX == 0) {
            // 128 rows x 32 floats: 1024 float4 -> 4 async b128 per thread
#pragma unroll
            for (int it = 0; it < 4; ++it) {
                const int i   = tid + it * 256;
                const int row = i >> 3;        // 8 float4 per LDS row
                const int c4  = i & 7;
                async_copy_b128(&ldsX[buf][row][c4 * 4],
                                Xb + (size_t)(m0 + row) * Kdim + k0 + c4 * 4);
            }
        } else {
            // tile(r,c) = X[b][k0+c][m0+r]; read 32 global rows of 128
            // contiguous floats (coalesced along m), scatter-transpose to LDS
#pragma unroll
            for (int it = 0; it < 4; ++it) {
                const int i    = tid + it * 256;
                const int krow = i >> 5;       // 32 float4 per global row
                const int m4   = i & 31;
                const float4 v4 = *(const float4*)(Xb + (size_t)(k0 + krow) * Mdim + m0 + m4 * 4);
                ldsX[buf][m4 * 4 + 0][krow] = v4.x;
                ldsX[buf][m4 * 4 + 1][krow] = v4.y;
                ldsX[buf][m4 * 4 + 2][krow] = v4.z;
                ldsX[buf][m4 * 4 + 3][krow] = v4.w;
            }
        }
    };
    auto stageW = [&](int k0, int buf) {
        // 32 k-rows x 32 r: 256 float4 -> 1 per thread, stored transposed
        const int row = tid >> 3;              // k row 0..31
        const int r4  = tid & 7;
        const float4 v4 = *(const float4*)(Wb + (size_t)(k0 + row) * R_ + r4 * 4);
        ldsWt[buf][r4 * 4 + 0][row] = v4.x;
        ldsWt[buf][r4 * 4 + 1][row] = v4.y;
        ldsWt[buf][r4 * 4 + 2][row] = v4.z;
        ldsWt[buf][r4 * 4 + 3][row] = v4.w;
    };

    // ---- software pipeline: prologue -------------------------------------
    stageX(0, 0);
    stageW(0, 0);
    wait_async0();
    __syncthreads();

    const int nch = Kdim / TK;
    for (int n = 0; n < nch; ++n) {
        const int cur = n & 1;
        const int nxt = cur ^ 1;
        if (n + 1 < nch) {          // fill next buffer while computing current
            stageX((n + 1) * TK, nxt);
            stageW((n + 1) * TK, nxt);
        }

        // ---- MAC: 8 WMMA steps over the K chunk (all operands ds_load_b64)
#pragma unroll
        for (int kk = 0; kk < TK; kk += 4) {
            // A 16x4 f32: lanes 0-15 hold K=kk,kk+1; lanes 16-31 hold K=kk+2,kk+3
            const v2f a  = *(const v2f*)&ldsX [cur][wr + l16][kk + half * 2];
            // B 4x16 f32: same K striping, N = lane&15 (two N-tiles: 0..15, 16..31)
            const v2f b0 = *(const v2f*)&ldsWt[cur][l16     ][kk + half * 2];
            const v2f b1 = *(const v2f*)&ldsWt[cur][16 + l16][kk + half * 2];
            c0 = __builtin_amdgcn_wmma_f32_16x16x4_f32(false, a, false, b0, (short)0, c0, false, false);
            c1 = __builtin_amdgcn_wmma_f32_16x16x4_f32(false, a, false, b1, (short)0, c1, false, false);
        }

        wait_async0();              // next buffer's async fills complete
        __syncthreads();            // all waves done reading cur / writing nxt
    }

    // ---- store D: VGPR j holds M = j + 8*half, N = lane&15 ---------------
    float* Ob = Out + (size_t)b * Mdim * R_;
#pragma unroll
    for (int j = 0; j < 8; ++j) {
        const int m = m0 + wr + j + half * 8;
        Ob[(size_t)m * R_ + l16]      = c0[j];
        Ob[(size_t)m * R_ + 16 + l16] = c1[j];
    }
}

// ---------------------------------------------------------------------------
// Gauss-Seidel column sweep (sequential in r, parallel over B x Mdim rows)
//   term2 = sum_k u[k]*G[k,r] - u[r]*G[r,r]
//   u[r]  = (soft(a[r]-term2, l1) + eps) / (G[r,r] + l2 + eps)
// ---------------------------------------------------------------------------
__global__ __launch_bounds__(256)
void gauss_seidel_kernel(const float* __restrict__ A, const float* __restrict__ G,
                         const float* __restrict__ Uin, float* __restrict__ Uout,
                         int Mdim, float l1, float l2)
{
    __shared__ float g[R_ * R_];
    const int b = blockIdx.y;
    const int m = blockIdx.x * 256 + threadIdx.x;

    for (int i = threadIdx.x; i < R_ * R_; i += 256)
        g[i] = G[(size_t)b * R_ * R_ + i];
    __syncthreads();

    const size_t base = ((size_t)b * Mdim + m) * R_;
    float ur[R_], ar[R_];
#pragma unroll
    for (int r = 0; r < R_; ++r) { ur[r] = Uin[base + r]; ar[r] = A[base + r]; }

    constexpr float eps = 1e-16f;
#pragma unroll
    for (int r = 0; r < R_; ++r) {
        const float brr = g[r * R_ + r];
        float t2 = 0.f;
#pragma unroll
        for (int k = 0; k < R_; ++k)
            t2 += ur[k] * g[k * R_ + r];
        t2 -= ur[r] * brr;
        const float z  = ar[r] - t2;
        const float az = fabsf(z) - l1;
        const float num = (az > 0.f) ? copysignf(az, z) : 0.f;
        ur[r] = (num + eps) / (brr + l2 + eps);
    }

#pragma unroll
    for (int r = 0; r < R_; ++r) Uout[base + r] = ur[r];
}

// ---------------------------------------------------------------------------
extern "C" void kernel_launch(void* const* d_in, const int* in_sizes, int n_in,
                              void* d_out, int out_size, void* d_ws, size_t ws_size,
                              hipStream_t stream)
{
    (void)in_sizes; (void)n_in; (void)out_size; (void)ws_size;

    const float* x = (const float*)d_in[0];   // [B, M, N]
    const float* u = (const float*)d_in[1];   // [B, M, R]
    const float* v = (const float*)d_in[2];   // [B, N, R]

    float* u_out = (float*)d_out;                         // [B, M, R]
    float* v_out = (float*)d_out + (size_t)B_ * M_ * R_;  // [B, N, R]

    float* a_buf = (float*)d_ws;                          // [B, max(M,N), R]  (2 MB)
    float* g_buf = a_buf + (size_t)B_ * M_ * R_;          // [B, R, R]         (32 KB)

    // L2=(0.01,0.01), L1_RATIO=0.5
    const float l1_u = 0.01f * 0.5f * (float)N_;
    const float l2_u = 0.01f * (1.0f - 0.5f) * (float)N_;
    const float l1_v = 0.01f * 0.5f * (float)M_;
    const float l2_v = 0.01f * (1.0f - 0.5f) * (float)M_;

    // ---- factor 0: update u ----
    gram32_kernel<<<B_, 1024, 0, stream>>>(v, g_buf, N_);
    gemm_xw_kernel<0><<<dim3(M_ / TM, B_), 256, 0, stream>>>(x, v, a_buf, M_, N_);
    gauss_seidel_kernel<<<dim3(M_ / 256, B_), 256, 0, stream>>>(a_buf, g_buf, u, u_out, M_, l1_u, l2_u);

    // ---- factor 1: update v (uses fresh u) ----
    gram32_kernel<<<B_, 1024, 0, stream>>>(u_out, g_buf, M_);
    gemm_xw_kernel<1><<<dim3(N_ / TM, B_), 256, 0, stream>>>(x, u_out, a_buf, N_, M_);
    gauss_seidel_kernel<<<dim3(N_ / 256, B_), 256, 0, stream>>>(a_buf, g_buf, v, v_out, N_, l1_v, l2_v);
}